// FaceKernelCorrelation_62826781605925
// MI455X (gfx1250) — compile-verified
//
#include <hip/hip_runtime.h>
#include <hip/hip_bf16.h>

typedef __attribute__((ext_vector_type(2))) float v2f;
typedef __attribute__((ext_vector_type(8))) float v8f;

#define SIGMA2_INV   25.0f                 // 1/sigma^2,  sigma=0.2
#define HALF_S2_INV  12.5f                 // 1/(2 sigma^2)
#define LOG2E        1.4426950408889634f
#define BN_EPS       1e-5f
#define BB 8
#define FF 16384
#define KK 3
#define MM 64

// swizzle immediates: group-of-32 xor swaps (offset[15]=0, xor_mask[14:10], and_mask[4:0]=0x1F)
#define SWAPX1 0x041F
#define SWAPX2 0x081F

template <int IMM>
__device__ __forceinline__ float swz(float v) {
    return __builtin_bit_cast(float,
        __builtin_amdgcn_ds_swizzle(__builtin_bit_cast(int, v), IMM));
}

// ---------------------------------------------------------------------------
// Kernel 0: build B-matrix columns and per-column scales from (alpha, beta).
// B pre-scaled by log2(e) so the hot loop is WMMA -> raw v_exp_f32 (exp2).
//   Bmat[c][col]  = w_c/sigma^2 * log2e        (c = 0..2)
//   Bmat[3][col]  = -1/(2 sigma^2) * log2e     (multiplies p2 carried in A)
//   wscale[col]   = exp(-||w||^2/(2 sigma^2)) / 16
// so  exp(dist/(-2 sigma^2))/16 = exp2(D) * wscale,  D = A x B.
// ---------------------------------------------------------------------------
__global__ void fkc_prep(const float* __restrict__ walpha,
                         const float* __restrict__ wbeta,
                         float* __restrict__ Bmat,
                         float* __restrict__ wscale) {
    int col = threadIdx.x;            // 0..255 = m*4 + j
    if (col < MM * 4) {
        float a  = walpha[col];
        float bt = wbeta[col];
        float sa = __sinf(a), ca = __cosf(a);
        float sb = __sinf(bt), cb = __cosf(bt);
        float w0 = sa * cb, w1 = sa * sb, w2 = ca;
        Bmat[0 * 256 + col] = w0 * SIGMA2_INV * LOG2E;
        Bmat[1 * 256 + col] = w1 * SIGMA2_INV * LOG2E;
        Bmat[2 * 256 + col] = w2 * SIGMA2_INV * LOG2E;
        Bmat[3 * 256 + col] = -HALF_S2_INV * LOG2E;
        wscale[col] = __builtin_amdgcn_exp2f(
            -(w0 * w0 + w1 * w1 + w2 * w2) * HALF_S2_INV * LOG2E - 4.0f);
    }
}

// ---------------------------------------------------------------------------
// Kernel 1: main correlation. One wave per 4 faces.
// Rows (16)  = 4 faces x 4 points (center + 3 gathered neighbors).
// Cols (256) = 64 channels x 4 kernel points, tiled 16 at a time.
// D = v_wmma_f32_16x16x4_f32(A, Bt, 0) ; elem = exp2(D) ; scale ; reduce 4x4
// blocks -> feat[b][m][f] (pre-BN) written to d_out.
// ---------------------------------------------------------------------------
__global__ void fkc_corr(const float* __restrict__ normals,   // [B][3][F]
                         const long long* __restrict__ nidx,  // [B][F][K]
                         const float* __restrict__ Bmat,      // [4][256]
                         const float* __restrict__ wscale,    // [256]
                         float* __restrict__ out) {           // [B][M][F]
    int tid  = blockIdx.x * blockDim.x + threadIdx.x;
    int wave = tid >> 5;
    int lane = tid & 31;
    int hl   = lane & 15;              // row index (A) / col-in-tile (B)

    int b     = wave >> 12;            // wave / (F/4)
    int fbase = (wave & 4095) << 2;    // 4 faces per wave

    // ---- Build A: row = hl -> face fbase + (hl>>2), point i = hl&3 ----
    int f = fbase + (hl >> 2);
    int i = hl & 3;
    long long g;
    if (i == 0) {
        g = f;
    } else {
        g = nidx[((long long)(b * FF + f)) * KK + (i - 1)];
    }
    const float* nb = normals + (long long)b * 3 * FF;
    float px = nb[g];
    float py = nb[FF + g];
    float pz = nb[2 * FF + g];
    float p2 = px * px + py * py + pz * pz;

    // A 16x4 f32 layout: v0 = K0 (lanes 0-15) / K2 (lanes 16-31), v1 = K1 / K3
    v2f A;
    A.x = (lane < 16) ? px : pz;
    A.y = (lane < 16) ? py : p2;

    float acc0[16];   // per-tile sums, face0 (lanes<16) / face2 (lanes>=16)
    float acc1[16];   // per-tile sums, face1 / face3

#pragma unroll
    for (int t = 0; t < 16; ++t) {
        int col = (t << 4) | hl;
        // B 4x16 f32 layout: v0 = row K0 (lanes 0-15) / K2 (lanes 16-31), v1 = K1 / K3
        v2f Bt;
        Bt.x = (lane < 16) ? Bmat[col] : Bmat[2 * 256 + col];
        Bt.y = (lane < 16) ? Bmat[256 + col] : Bmat[3 * 256 + col];
        float ws = wscale[col];

        v8f C = {};
        v8f D = __builtin_amdgcn_wmma_f32_16x16x4_f32(
            false, A, false, Bt, (short)0, C, false, false);

        // 8 independent exps first, then a balanced add tree (no trans-dep stalls)
        float x0 = __builtin_amdgcn_exp2f(D[0]);
        float x1 = __builtin_amdgcn_exp2f(D[1]);
        float x2 = __builtin_amdgcn_exp2f(D[2]);
        float x3 = __builtin_amdgcn_exp2f(D[3]);
        float x4 = __builtin_amdgcn_exp2f(D[4]);
        float x5 = __builtin_amdgcn_exp2f(D[5]);
        float x6 = __builtin_amdgcn_exp2f(D[6]);
        float x7 = __builtin_amdgcn_exp2f(D[7]);
        float s01 = x0 + x1, s23 = x2 + x3;
        float s45 = x4 + x5, s67 = x6 + x7;
        // column scale applied once per row-group (exp factorization)
        float e0 = (s01 + s23) * ws;       // rows 0-3: face0 / rows 8-11: face2
        float e1 = (s45 + s67) * ws;       // rows 4-7: face1 / rows 12-15: face3

        // reduce the 4 kernel-point columns (j) within each channel m
        e0 += swz<SWAPX1>(e0);
        e0 += swz<SWAPX2>(e0);
        e1 += swz<SWAPX1>(e1);
        e1 += swz<SWAPX2>(e1);

        acc0[t] = e0;
        acc1[t] = e1;
    }

    // one divergent store block instead of 16 exec toggles
    if ((lane & 3) == 0) {
        int mbase = hl >> 2;
        int face0 = fbase + ((lane < 16) ? 0 : 2);
        int face1 = fbase + ((lane < 16) ? 1 : 3);
#pragma unroll
        for (int t = 0; t < 16; ++t) {
            int m = (t << 2) | mbase;
            out[(((b << 6) | m) << 14) | face0] = acc0[t];
            out[(((b << 6) | m) << 14) | face1] = acc1[t];
        }
    }
}

// ---------------------------------------------------------------------------
// Kernel 2: per-channel batch stats (mean, rstd) over (B, F). One block per m.
// float4-vectorized loads.
// ---------------------------------------------------------------------------
__global__ void fkc_bnstats(const float* __restrict__ feat,
                            float* __restrict__ stats) {
    __shared__ float ssum[256];
    __shared__ float ssq[256];
    int m   = blockIdx.x;
    int tid = threadIdx.x;
    float s = 0.f, s2 = 0.f;
    // 8*16384/4 = 32768 quads per channel
    for (int q = tid; q < (BB * FF) / 4; q += 256) {
        int b  = q >> 12;                 // 4096 quads per (b, m) row
        int f4 = q & 4095;
        const float4 v = *reinterpret_cast<const float4*>(
            feat + ((((b << 6) | m) << 14) | (f4 << 2)));
        s  += v.x + v.y + v.z + v.w;
        s2 += v.x * v.x + v.y * v.y + v.z * v.z + v.w * v.w;
    }
    ssum[tid] = s;
    ssq[tid]  = s2;
    __syncthreads();
    for (int off = 128; off > 0; off >>= 1) {
        if (tid < off) {
            ssum[tid] += ssum[tid + off];
            ssq[tid]  += ssq[tid + off];
        }
        __syncthreads();
    }
    if (tid == 0) {
        const float N = (float)(BB * FF);
        float mean = ssum[0] / N;
        float var  = ssq[0] / N - mean * mean;
        stats[2 * m]     = mean;
        stats[2 * m + 1] = rsqrtf(var + BN_EPS);
    }
}

// ---------------------------------------------------------------------------
// Kernel 3: normalize + affine + ReLU, in place, float4 vectorized.
// ---------------------------------------------------------------------------
__global__ void fkc_bnapply(float* __restrict__ out,
                            const float* __restrict__ stats,
                            const float* __restrict__ gamma,
                            const float* __restrict__ beta) {
    int q = blockIdx.x * blockDim.x + threadIdx.x;   // quad index
    int i = q << 2;                                   // element index
    int m = (i >> 14) & (MM - 1);
    float mean = stats[2 * m];
    float gs   = gamma[m] * stats[2 * m + 1];
    float bt   = beta[m];
    float4 v = *reinterpret_cast<float4*>(out + i);
    v.x = fmaxf((v.x - mean) * gs + bt, 0.f);
    v.y = fmaxf((v.y - mean) * gs + bt, 0.f);
    v.z = fmaxf((v.z - mean) * gs + bt, 0.f);
    v.w = fmaxf((v.w - mean) * gs + bt, 0.f);
    *reinterpret_cast<float4*>(out + i) = v;
}

// ---------------------------------------------------------------------------
extern "C" void kernel_launch(void* const* d_in, const int* in_sizes, int n_in,
                              void* d_out, int out_size, void* d_ws, size_t ws_size,
                              hipStream_t stream) {
    const float*     normals = (const float*)d_in[0];      // [8][3][16384]
    const float*     walpha  = (const float*)d_in[1];      // [1][64][4]
    const float*     wbeta   = (const float*)d_in[2];      // [1][64][4]
    const float*     gamma   = (const float*)d_in[3];      // [64]
    const float*     beta    = (const float*)d_in[4];      // [64]
    const long long* nidx    = (const long long*)d_in[5];  // [8][16384][3] int64
    float* out = (float*)d_out;                            // [8][64][16384]

    float* ws     = (float*)d_ws;
    float* Bmat   = ws;            // 1024 floats
    float* wscale = ws + 1024;     // 256 floats
    float* stats  = ws + 1280;     // 128 floats

    fkc_prep<<<1, 256, 0, stream>>>(walpha, wbeta, Bmat, wscale);

    // one wave per 4 faces: B*F/4 = 32768 waves, 8 waves per block
    fkc_corr<<<4096, 256, 0, stream>>>(normals, nidx, Bmat, wscale, out);

    fkc_bnstats<<<MM, 256, 0, stream>>>(out, stats);

    // 8*64*16384 / 4 = 2097152 quads, 256 threads/block
    fkc_bnapply<<<8192, 256, 0, stream>>>(out, stats, gamma, beta);
}